// SingleRenderer_41463614275763
// MI455X (gfx1250) — compile-verified
//
#include <hip/hip_runtime.h>
#include <math.h>

typedef float v2f __attribute__((ext_vector_type(2)));
typedef float v8f __attribute__((ext_vector_type(8)));

#define N_RAYS   8192
#define N_INIT   512
#define N_UPS    512
#define N_IMP    64
#define MAX_IT   3
#define MAX_BIS  10
#define EPS_F    0.1f
#define ALPHA_N  10.0f
#define BETA_N   0.1f
#define FAR_F    6.0f
#define SPH_R    1.5f
#define MMAX     2048
#define MPAD     (MMAX + 16)

__device__ __forceinline__ int imin(int a, int b) { return a < b ? a : b; }
__device__ __forceinline__ int imax(int a, int b) { return a > b ? a : b; }

__device__ __forceinline__ float wave_max32(float v) {
  #pragma unroll
  for (int o = 16; o; o >>= 1) v = fmaxf(v, __shfl_xor(v, o, 32));
  return v;
}

// ---------------------------------------------------------------------------
// Inclusive prefix sum over 256-element tiles in LDS using the matrix pipe.
// A 16x16 tile of 16 segments is scanned with a chain of 4 accumulating
// V_WMMA_F32_16X16X4_F32 ops: D = sum_g A_g(16x4 triangular mask) @ B_g(4x16
// data rows).  Cross-segment / cross-tile offsets via wave32 shuffles.
// EXEC is all-ones at every WMMA (no divergence in this function's WMMA path).
// ---------------------------------------------------------------------------
__device__ void scan256_wmma(float* arr, int ntiles, int lane) {
  const int half = lane >> 4;   // 0: rows 0-7 / K 0,1   1: rows 8-15 / K 2,3
  const int l15  = lane & 15;
  float carry = 0.f;
  for (int t = 0; t < ntiles; ++t) {
    float* base = arr + (t << 8);
    union { v8f v; float f[8]; } acc;
    #pragma unroll
    for (int r = 0; r < 8; ++r) acc.f[r] = 0.f;
    #pragma unroll
    for (int g = 0; g < 4; ++g) {
      const int k0 = 4 * g + 2 * half;        // K indices this lane supplies
      v2f a, b;
      a.x = (k0     <= l15) ? 1.f : 0.f;      // L16[m=l15, k0]
      a.y = (k0 + 1 <= l15) ? 1.f : 0.f;      // L16[m=l15, k0+1]
      b.x = base[(l15 << 4) + k0];            // X[k0,   n=l15]
      b.y = base[(l15 << 4) + k0 + 1];        // X[k0+1, n=l15]
      acc.v = __builtin_amdgcn_wmma_f32_16x16x4_f32(
          false, a, false, b, (short)0, acc.v, false, false);
    }
    // segment totals live in row 15 -> acc.f[7] of lanes 16..31
    float tot = __shfl(acc.f[7], 16 + l15, 32);
    float x = tot;                            // per-half inclusive scan of 16 totals
    #pragma unroll
    for (int o = 1; o < 16; o <<= 1) {
      float y = __shfl_up(x, o, 32);
      if (l15 >= o) x += y;
    }
    const float segoff = carry + (x - tot);   // exclusive offset for segment l15
    #pragma unroll
    for (int r = 0; r < 8; ++r)               // Y[m,n] -> base[16n + m]
      base[(l15 << 4) + (half << 3) + r] = acc.f[r] + segoff;
    carry += __shfl(x, 15, 32);               // tile total
  }
}

// ---------------------------------------------------------------------------
// max over error_bound(d, sdf, alpha, beta); optionally writes bounds array.
// bnd_mode: 0 = none, 1 = raw (nan->inf), 2 = clipped to [0, 1e5]
// ---------------------------------------------------------------------------
__device__ float error_bound_max(const float* s_d, const float* s_sdf, int M,
                                 float alpha, float beta,
                                 float* s_w, float* s_e,
                                 float* s_bnd, int bnd_mode, int lane) {
  const int n    = M - 1;
  const int npad = (n + 255) & ~255;
  const float ib = 1.f / beta;
  const float ce = alpha * 0.25f * ib;
  for (int i = lane; i < npad; i += 32) {
    float w = 0.f, er = 0.f;
    if (i < n) {
      const float sdf0 = s_sdf[i], sdf1 = s_sdf[i + 1];
      const float d0 = s_d[i], d1 = s_d[i + 1];
      const float a0 = fabsf(sdf0), a1 = fabsf(sdf1);
      const float e0 = 0.5f * __expf(-a0 * ib);
      const float psi = (sdf0 >= 0.f) ? e0 : (1.f - e0);
      const float delta = d1 - d0;
      w = alpha * psi * delta;                          // sigma*delta
      const float dstar = fmaxf(0.5f * (a0 + a1 - delta), 0.f);
      er = ce * delta * delta * __expf(-dstar * ib);
    }
    s_w[i] = w;
    s_e[i] = er;
  }
  scan256_wmma(s_w, npad >> 8, lane);   // inclusive cumsum(sigma*delta)
  scan256_wmma(s_e, npad >> 8, lane);   // inclusive cumsum(errs)
  float m = -3.4e38f;
  for (int i = lane; i < n; i += 32) {
    const float Rt = (i == 0) ? 0.f : s_w[i - 1];
    float b = __expf(-Rt) * (__expf(s_e[i]) - 1.f);
    if (!(b == b)) b = __builtin_inff();                // nan -> inf
    if (bnd_mode == 1)      s_bnd[i] = b;
    else if (bnd_mode == 2) s_bnd[i] = fminf(fmaxf(b, 0.f), 1e5f);
    m = fmaxf(m, b);
  }
  return wave_max32(m);
}

// ---------------------------------------------------------------------------
// opacity_invert_cdf_sample -> 64 importance samples written to outp
// ---------------------------------------------------------------------------
__device__ void opacity_sample(const float* s_d, const float* s_sdf, int M,
                               float alpha, float beta,
                               float* s_w, float* s_e, float* outp, int lane) {
  const int n    = M - 1;
  const int npad = (n + 255) & ~255;
  const float ib = 1.f / beta;
  for (int i = lane; i < npad; i += 32) {
    float w = 0.f;
    if (i < n) {
      const float sdf0 = s_sdf[i];
      const float e0 = 0.5f * __expf(-fabsf(sdf0) * ib);
      const float psi = (sdf0 >= 0.f) ? e0 : (1.f - e0);
      w = alpha * psi * (s_d[i + 1] - s_d[i]);
    }
    s_w[i] = w;
  }
  scan256_wmma(s_w, npad >> 8, lane);
  for (int i = lane; i < n; i += 32) {
    const float Rt = (i == 0) ? 0.f : s_w[i - 1];
    s_e[i] = 1.f - __expf(-Rt);                          // opacity (monotone)
  }
  const float last = s_e[n - 1];
  for (int j = lane; j < N_IMP; j += 32) {
    const float u = (float)j * (1.f / (float)(N_IMP - 1)) * last;
    int lo = 0, hi = n;                                  // searchsorted right
    while (lo < hi) { int mid = (lo + hi) >> 1; if (s_e[mid] <= u) lo = mid + 1; else hi = mid; }
    const int below = imin(imax(lo - 1, 0), n - 1);
    const int above = imin(lo, n - 1);
    const float c0 = s_e[below], c1 = s_e[above];
    float den = c1 - c0; if (den < 1e-5f) den = 1.f;
    const float tt = (u - c0) / den;
    outp[j] = fmaf(tt, s_d[above] - s_d[below], s_d[below]);
  }
}

// ---------------------------------------------------------------------------
// sample_pdf over bounds -> 512 new (sorted) depths into s_up
// ---------------------------------------------------------------------------
__device__ void upsample_pdf(const float* s_d, int M, const float* s_bnd,
                             float* s_w, float* s_e, float* s_up, int lane) {
  const int n    = M - 1;
  const int npad = (n + 255) & ~255;
  for (int i = lane; i < npad; i += 32) s_w[i] = (i < n) ? s_bnd[i] : 0.f;
  scan256_wmma(s_w, npad >> 8, lane);
  const float invTot = 1.f / (s_w[n - 1] + 1e-5f);
  for (int i = lane; i < M; i += 32)                      // cdf, length M
    s_e[i] = (i == 0) ? 0.f : s_w[i - 1] * invTot;
  const float du = 1.f / (float)(N_UPS + 1);              // u = j/513, j=1..512
  for (int j = lane; j < N_UPS; j += 32) {
    const float u = (float)(j + 1) * du;
    int lo = 0, hi = M;
    while (lo < hi) { int mid = (lo + hi) >> 1; if (s_e[mid] <= u) lo = mid + 1; else hi = mid; }
    const int below = imin(imax(lo - 1, 0), M - 1);
    const int above = imin(lo, M - 1);
    const float c0 = s_e[below], c1 = s_e[above];
    float den = c1 - c0; if (den < 1e-5f) den = 1.f;
    const float tt = (u - c0) / den;
    s_up[j] = fmaf(tt, s_d[above] - s_d[below], s_d[below]);
  }
}

// ---------------------------------------------------------------------------
// One wave32 per ray; whole state LDS-resident (~43 KB << 320 KB per WGP).
// ---------------------------------------------------------------------------
__global__ __launch_bounds__(32)
void fine_sample_kernel(const float* __restrict__ init_dvals,
                        const float* __restrict__ rays_o,
                        const float* __restrict__ rays_d,
                        float* __restrict__ out) {
  __shared__ float s_d[MPAD], s_sdf[MPAD], s_w[MPAD], s_e[MPAD], s_bnd[MPAD];
  __shared__ float s_up[N_UPS + 4];

  const int ray  = blockIdx.x;
  const int lane = threadIdx.x;

  const float ox = rays_o[3 * ray], oy = rays_o[3 * ray + 1], oz = rays_o[3 * ray + 2];
  const float dx = rays_d[3 * ray], dy = rays_d[3 * ray + 1], dz = rays_d[3 * ray + 2];
  const float qa = dx * dx + dy * dy + dz * dz;
  const float qb = 2.f * (ox * dx + oy * dy + oz * dz);
  const float qc = ox * ox + oy * oy + oz * oz;

  int M = N_INIT;
  for (int i = lane; i < N_INIT; i += 32) {
    const float t = init_dvals[(size_t)ray * N_INIT + i];
    s_d[i]   = t;
    s_sdf[i] = sqrtf(fmaf(t, fmaf(t, qa, qb), qc)) - SPH_R;   // sphere SDF
  }

  float beta = sqrtf(FAR_F * FAR_F / (4.f * (float)(N_INIT - 1) * logf(1.f + EPS_F)));
  float* fineout = out + (size_t)ray * N_IMP;

  float nbm = error_bound_max(s_d, s_sdf, M, ALPHA_N, BETA_N, s_w, s_e, nullptr, 0, lane);
  bool  converged  = !(nbm > EPS_F);
  float iter_usage = 0.f;
  float beta_out   = BETA_N;

  if (converged) {
    opacity_sample(s_d, s_sdf, M, ALPHA_N, BETA_N, s_w, s_e, fineout, lane);
  } else {
    // initial bounds with (1/beta, beta), unclipped
    error_bound_max(s_d, s_sdf, M, 1.f / beta, beta, s_w, s_e, s_bnd, 1, lane);

    for (int it = 1; it <= MAX_IT; ++it) {
      upsample_pdf(s_d, M, s_bnd, s_w, s_e, s_up, lane);

      // merge-path merge of sorted s_d (M) with sorted s_up (512) into s_e
      {
        const int nb    = N_UPS;
        const int Mout  = M + nb;
        const int chunk = Mout >> 5;                // per-lane output span
        const int k0    = lane * chunk;
        int ilo = imax(0, k0 - nb), ihi = imin(k0, M);
        while (ilo < ihi) {                          // co-rank binary search
          const int i = (ilo + ihi) >> 1;
          const int j = k0 - 1 - i;
          if (s_d[i] <= s_up[j]) ilo = i + 1; else ihi = i;
        }
        int i = ilo, j = k0 - ilo;
        for (int t = 0; t < chunk; ++t) {
          const bool takeA = (j >= nb) || (i < M && s_d[i] <= s_up[j]);
          float v;
          if (takeA) v = s_d[i++]; else v = s_up[j++];
          s_e[k0 + t] = v;
        }
        for (int i2 = lane; i2 < Mout; i2 += 32) {   // copy back + fresh SDF
          const float t = s_e[i2];
          s_d[i2]   = t;
          s_sdf[i2] = sqrtf(fmaf(t, fmaf(t, qa, qb), qc)) - SPH_R;
        }
        M = Mout;
      }

      nbm = error_bound_max(s_d, s_sdf, M, ALPHA_N, BETA_N, s_w, s_e, nullptr, 0, lane);
      if (!(nbm > EPS_F)) {
        converged  = true;
        iter_usage = (float)it;
        opacity_sample(s_d, s_sdf, M, ALPHA_N, BETA_N, s_w, s_e, fineout, lane);
        break;
      }

      // bisection: smallest beta meeting the error bound (wave-uniform scalars)
      float bl = BETA_N, br = beta;
      for (int b = 0; b < MAX_BIS; ++b) {
        const float bt = 0.5f * (bl + br);
        const float bm = error_bound_max(s_d, s_sdf, M, 1.f / bt, bt,
                                         s_w, s_e, nullptr, 0, lane);
        if (bm <= EPS_F) br = bt; else bl = bt;
      }
      beta = br;
      if (it < MAX_IT)   // bounds for next upsample, clipped to [0, 1e5]
        error_bound_max(s_d, s_sdf, M, 1.f / beta, beta, s_w, s_e, s_bnd, 2, lane);
    }

    if (!converged) {
      opacity_sample(s_d, s_sdf, M, 1.f / beta, beta, s_w, s_e, fineout, lane);
      iter_usage = -1.f;
      beta_out   = beta;
    }
  }

  if (lane == 0) {
    out[(size_t)N_RAYS * N_IMP + ray]          = beta_out;    // beta_out
    out[(size_t)N_RAYS * N_IMP + N_RAYS + ray] = iter_usage;  // iter_usage
  }
}

extern "C" void kernel_launch(void* const* d_in, const int* in_sizes, int n_in,
                              void* d_out, int out_size, void* d_ws, size_t ws_size,
                              hipStream_t stream) {
  (void)in_sizes; (void)n_in; (void)out_size; (void)d_ws; (void)ws_size;
  const float* init_dvals = (const float*)d_in[0];
  const float* rays_o     = (const float*)d_in[1];
  const float* rays_d     = (const float*)d_in[2];
  float* out = (float*)d_out;
  fine_sample_kernel<<<N_RAYS, 32, 0, stream>>>(init_dvals, rays_o, rays_d, out);
}